// ECELoss_38706245272183
// MI455X (gfx1250) — compile-verified
//
#include <hip/hip_runtime.h>

// ECE loss, MI455X (gfx1250). Bandwidth-bound streaming reduction:
// ~336 MB read-once @ 23.3 TB/s => ~14.5 us floor. No matrix structure => no
// WMMA (AI ~1.4 flop/byte; matrix pipes cannot remove any bytes here).
// Data path: coalesced non-temporal b32 loads (stream-once, keep L2 clean),
// register-resident 19-channel softmax-max (score = 1/sum(exp2((v-m)*log2e))),
// per-thread private LDS histograms (unique address per lane -> ds_add with
// zero contention), packed u32 count|correct, guarded gfx1250 prefetch.

#define ECE_NPIX (8u * 512u * 1024u)   // 4,194,304
#define ECE_HW   (512u * 1024u)        // 524,288 (= 2^19)
#define ECE_C    19
#define ECE_NB   10
#define ECE_TPB  256
#define ECE_GRID 2048                  // 2048*256*8 == NPIX exactly

__global__ __launch_bounds__(32) void ece_init_kernel(float* __restrict__ ws) {
    int t = threadIdx.x;
    if (t < 32) ws[t] = 0.0f;
}

__global__ __launch_bounds__(ECE_TPB) void ece_hist_kernel(
        const float* __restrict__ logits,
        const int*   __restrict__ target,
        float* __restrict__ ws) {
    // Private per-thread histograms: 256 threads x 10 bins.
    //   sConf: float confidence sums
    //   sPk:   u32 packed (correct << 16) | count  (per-thread incr <= 8,
    //          per-block totals <= 2048 per field -> no cross-field carry)
    // 2560*4B*2 = 20 KB (<< 320 KB/WGP LDS).
    __shared__ float    sConf[ECE_TPB * ECE_NB];
    __shared__ unsigned sPk  [ECE_TPB * ECE_NB];

    const int tid = threadIdx.x;
#pragma unroll
    for (int j = 0; j < ECE_NB; ++j) {
        sConf[tid * ECE_NB + j] = 0.0f;
        sPk  [tid * ECE_NB + j] = 0u;
    }
    __syncthreads();

    const unsigned gsize = gridDim.x * blockDim.x;   // == ECE_HW for this launch
    const float K = 1.44269504088896340736f;         // log2(e)

    for (unsigned pix = blockIdx.x * blockDim.x + threadIdx.x;
         pix < ECE_NPIX; pix += gsize) {
        const unsigned b  = pix >> 19;               // HW = 2^19
        const unsigned hw = pix & (ECE_HW - 1u);
        const float* p = logits + (size_t)b * (size_t)(ECE_C * ECE_HW) + hw;

        // Prefetch next iteration's channel-0 line (same hw, b+1). Guarded:
        // WGP-scope prefetch is non-speculative, so never touch past the end.
        if (pix + gsize < ECE_NPIX)
            __builtin_prefetch(p + (size_t)(ECE_C) * ECE_HW, 0, 0);

        float v[ECE_C];
#pragma unroll
        for (int c = 0; c < ECE_C; ++c)
            v[c] = __builtin_nontemporal_load(p + (size_t)c * ECE_HW);
        const int tl = __builtin_nontemporal_load(target + pix);

        // pass 1: running max (v_max_f32 x18)
        float m = v[0];
#pragma unroll
        for (int c = 1; c < ECE_C; ++c) m = fmaxf(m, v[c]);

        // pass 2 (descending c => final write = smallest c = first occurrence):
        // fma + v_exp_f32 + add + cmp + cndmask per channel
        const float mk = m * K;
        float s = 0.0f;
        int lab = 0;
#pragma unroll
        for (int c = ECE_C - 1; c >= 0; --c) {
            s += __builtin_amdgcn_exp2f(__builtin_fmaf(v[c], K, -mk));
            lab = (v[c] == m) ? c : lab;
        }

        const float score = 1.0f / s;                // IEEE div, s in [1,19]
        // bin i covers (i/10, (i+1)/10]  (searchsorted left - 1, clamped)
        int bin = (int)__builtin_ceilf(score * 10.0f) - 1;
        bin = bin < 0 ? 0 : (bin > ECE_NB - 1 ? ECE_NB - 1 : bin);

        const int idx = tid * ECE_NB + bin;          // unique per lane
        atomicAdd(&sConf[idx], score);
        atomicAdd(&sPk[idx], 1u + ((lab == tl) ? (1u << 16) : 0u));
    }
    __syncthreads();

    // Tree-reduce 256 private rows down to row 0 (once per block).
    for (int off = ECE_TPB / 2; off >= 1; off >>= 1) {
        if (tid < off) {
#pragma unroll
            for (int j = 0; j < ECE_NB; ++j) {
                sConf[tid * ECE_NB + j] += sConf[(tid + off) * ECE_NB + j];
                sPk  [tid * ECE_NB + j] += sPk  [(tid + off) * ECE_NB + j];
            }
        }
        __syncthreads();
    }

    // ws[0..9]=conf_sum, ws[10..19]=count, ws[20..29]=correct
    if (tid < ECE_NB) {
        atomicAdd(&ws[tid], sConf[tid]);
    } else if (tid < 2 * ECE_NB) {
        const unsigned pk = sPk[tid - ECE_NB];
        atomicAdd(&ws[tid], (float)(pk & 0xFFFFu));            // count
    } else if (tid < 3 * ECE_NB) {
        const unsigned pk = sPk[tid - 2 * ECE_NB];
        atomicAdd(&ws[tid], (float)(pk >> 16));                // correct
    }
}

__global__ __launch_bounds__(32) void ece_final_kernel(
        const float* __restrict__ ws, float* __restrict__ out) {
    if (threadIdx.x == 0) {
        float tot = 0.0f;
#pragma unroll
        for (int b = 0; b < ECE_NB; ++b) tot += ws[ECE_NB + b];
        float ece = 0.0f;
#pragma unroll
        for (int b = 0; b < ECE_NB; ++b) {
            const float n   = ws[ECE_NB + b];
            const float inv = 1.0f / (n + 1e-13f);
            const float acc = ws[2 * ECE_NB + b] * inv;
            const float cf  = ws[b] * inv;
            const float d   = fabsf(acc - cf);
            ece += d * d * (n / tot);
        }
        out[0] = ece;
    }
}

extern "C" void kernel_launch(void* const* d_in, const int* in_sizes, int n_in,
                              void* d_out, int out_size, void* d_ws, size_t ws_size,
                              hipStream_t stream) {
    (void)in_sizes; (void)n_in; (void)out_size; (void)ws_size;
    const float* logits = (const float*)d_in[0];
    const int*   target = (const int*)d_in[1];   // JAX default demotes int64->int32
    float*       ws     = (float*)d_ws;
    float*       out    = (float*)d_out;

    ece_init_kernel <<<1, 32, 0, stream>>>(ws);
    ece_hist_kernel <<<ECE_GRID, ECE_TPB, 0, stream>>>(logits, target, ws);
    ece_final_kernel<<<1, 32, 0, stream>>>(ws, out);
}